// Beam_Search_Tree_33586644254817
// MI455X (gfx1250) — compile-verified
//
#include <hip/hip_runtime.h>
#include <hip/hip_bf16.h>

// ---------------------------------------------------------------------------
// Beam-search tree: B=131072 rows, packed real GEMM (B x 128) * (128 x 252)
// via V_WMMA_F32_16X16X4_F32, then per-row products of pairwise sigmoids.
// v3: manual ping-pong double buffer with STATIC array indices (v2's dynamic
//     bb[cur] index spilled to scratch). 512-thread blocks, 128 KB LDS reused
//     for gain staging.
// ---------------------------------------------------------------------------

typedef __attribute__((ext_vector_type(2))) float v2f;
typedef __attribute__((ext_vector_type(8))) float v8f;

#define N_ANT   64
#define NCOLS   252            // real columns (126 complex)
#define NCOLP   256            // padded to 16 N-tiles
#define KSTEPS  32             // 128 / 4
#define NTILES  16             // 256 / 16
#define LDSF_B  (KSTEPS * NTILES * 32 * 2)   // 32768 floats = 128 KB frag buffer
#define LDSF_G  (16 * 128)                   // per-wave gain floats (16 rows x 128)
#define LDS_BYTES (LDSF_B * 4)               // 131072 B; gains reuse same region

// One real-packed weight value: Bmat[r][col], r in 0..127, col in 0..255.
// col = 2c + part: part 0 -> y_re combo column [Wr ; -Wi], part 1 -> [Wi ; Wr].
__device__ __forceinline__ float bval(int r, int col,
                                      const float* t0, const float* t1,
                                      const float* t2, const float* t3,
                                      const float* t4) {
  if (col >= NCOLS) return 0.0f;
  int c    = col >> 1;      // complex column 0..125
  int part = col & 1;
  int a    = r & 63;        // antenna index
  bool lower = (r < 64);    // r<64 -> h_real partition, else h_imag partition
  float wr, wi;
  if (c < 62) {
    // node layers: complex col = off_l + 2*node + child
    const float* th; int off;
    if      (c < 2)  { th = t0; off = 0;  }
    else if (c < 6)  { th = t1; off = 2;  }
    else if (c < 14) { th = t2; off = 6;  }
    else if (c < 30) { th = t3; off = 14; }
    else             { th = t4; off = 30; }
    int idx   = c - off;
    int node  = idx >> 1;
    int child = idx & 1;
    float theta = th[node * 128 + a * 2 + child]; // theta[node][a][child]
    __sincosf(theta, &wi, &wr);
    wr *= 0.125f; wi *= 0.125f;
  } else {
    // leaf DFT codebook: A[a,beam] = exp(j*pi*a*t_beam)/8,
    // t_beam = linspace(1, cos(pi-1e-6), 64)[beam]
    int beam = c - 62;
    const float cend = -0.99999999999950000f;  // cos(pi - 1e-6)
    float t = 1.0f + (float)beam * ((cend - 1.0f) * (1.0f / 63.0f));
    sincospif((float)a * t, &wi, &wr);
    wr *= 0.125f; wi *= 0.125f;
  }
  if (part == 0) return lower ? wr : -wi;
  else           return lower ? wi :  wr;
}

// Build weights directly in WMMA B-fragment layout:
//   frag[((k*16 + n)*32 + lane)] = { Bmat[4k+2*hi][16n+lml], Bmat[4k+1+2*hi][16n+lml] }
// where hi = lane>>4, lml = lane&15.  One float2 per (k,n,lane).
__global__ void build_weights_kernel(const float* __restrict__ t0,
                                     const float* __restrict__ t1,
                                     const float* __restrict__ t2,
                                     const float* __restrict__ t3,
                                     const float* __restrict__ t4,
                                     float2* __restrict__ wfrag) {
  int t = blockIdx.x * blockDim.x + threadIdx.x;   // 0 .. 16383
  if (t >= KSTEPS * NTILES * 32) return;
  int lane = t & 31;
  int n    = (t >> 5) & 15;
  int k    = t >> 9;
  int col  = (n << 4) | (lane & 15);
  int r0   = 4 * k + 2 * (lane >> 4);
  float b0 = bval(r0,     col, t0, t1, t2, t3, t4);
  float b1 = bval(r0 + 1, col, t0, t1, t2, t3, t4);
  wfrag[t] = make_float2(b0, b1);
}

// Main kernel: 16 waves / block, 16 batch rows per wave (256 rows / block).
__global__ void __launch_bounds__(512, 1)
beam_tree_kernel(const float* __restrict__ x,      // (B, 128)
                 float* __restrict__ out,          // (B, 64)
                 const float2* __restrict__ wfrag) // frag-layout weights
{
  extern __shared__ float lds[];
  const int tid = threadIdx.x;

  // --- 1. cooperative load of the 128 KB fragment matrix into LDS ---------
  {
    float4* dst = (float4*)lds;
    const float4* src = (const float4*)wfrag;
#pragma unroll
    for (int i = 0; i < 16; ++i)
      dst[i * 512 + tid] = src[i * 512 + tid];
  }
  __syncthreads();

  const int wave = tid >> 5;
  const int lane = tid & 31;
  const int hi   = lane >> 4;    // 0: rows M 0..7 / K even-pair lo, 1: hi
  const int lml  = lane & 15;
  const long rowBase = (long)blockIdx.x * 256 + wave * 16;

  // A-fragment source: lane L covers batch row (rowBase + lml), K pair 4k+2*hi
  const float* xrow = x + (rowBase + lml) * 128 + 2 * hi;

  // --- 2. WMMA GEMM: D(16x256) = Xtile(16x128) * Bmat(128x256) ------------
  v8f acc[NTILES];
#pragma unroll
  for (int n = 0; n < NTILES; ++n)
#pragma unroll
    for (int r = 0; r < 8; ++r) acc[n][r] = 0.0f;

  const v2f* bfr = (const v2f*)lds;

  // Manual ping-pong: two named register buffers, ALL indices static.
  v2f ba[NTILES], bpp[NTILES];
  v2f Aa = *(const v2f*)(xrow);
#pragma unroll
  for (int n = 0; n < NTILES; ++n) ba[n] = bfr[n * 32 + lane];

  for (int k = 0; k < KSTEPS; k += 2) {
    // prefetch step k+1 into the pong buffer
    v2f Ab = *(const v2f*)(xrow + 4 * (k + 1));
#pragma unroll
    for (int n = 0; n < NTILES; ++n)
      bpp[n] = bfr[((k + 1) * NTILES + n) * 32 + lane];

    // compute step k from the ping buffer
#pragma unroll
    for (int n = 0; n < NTILES; ++n)
      acc[n] = __builtin_amdgcn_wmma_f32_16x16x4_f32(
          false, Aa, false, ba[n], (short)0, acc[n], false, false);

    // prefetch step k+2 into the ping buffer (uniform guard)
    if (k + 2 < KSTEPS) {
      Aa = *(const v2f*)(xrow + 4 * (k + 2));
#pragma unroll
      for (int n = 0; n < NTILES; ++n)
        ba[n] = bfr[((k + 2) * NTILES + n) * 32 + lane];
    }

    // compute step k+1 from the pong buffer
#pragma unroll
    for (int n = 0; n < NTILES; ++n)
      acc[n] = __builtin_amdgcn_wmma_f32_16x16x4_f32(
          false, Ab, false, bpp[n], (short)0, acc[n], false, false);
  }

  // all waves done reading B fragments -> reuse the LDS region for gains
  __syncthreads();

  // --- 3. gains = y_re^2 + y_im^2 -> LDS staging (per-wave region) --------
  // D layout: acc[n][r] at lane L = d[row = r + 8*hi][col = 16n + lml]
  // complex pair (2c, 2c+1) sits in adjacent lanes -> shfl_xor(.,1)
  float* g = lds + wave * LDSF_G;
#pragma unroll
  for (int n = 0; n < NTILES; ++n) {
#pragma unroll
    for (int r = 0; r < 8; ++r) {
      float d  = acc[n][r];
      float dp = __shfl_xor(d, 1, 32);
      if ((lane & 1) == 0) {
        int ccol = 8 * n + (lml >> 1);       // complex column 0..127
        int row  = r + 8 * hi;               // tile row 0..15
        g[row * 128 + ccol] = d * d + dp * dp;
      }
    }
  }
  __syncthreads();

  // --- 4. tree products: out[j] = prod_l sigmoid(G[c_l] - G[c_l ^ 1]) -----
  // c_l = off[l] + (j >> (5 - l)), off = {0,2,6,14,30,62}
  const int prow = lane & 15;                // row within wave tile
  const int jb   = (lane >> 4) * 32;         // output half handled by lane
  const float* gr = g + prow * 128;
  float* orow = out + (rowBase + prow) * 64;
  const int off0 = 0, off1 = 2, off2 = 6, off3 = 14, off4 = 30, off5 = 62;
  for (int i = 0; i < 32; ++i) {
    int j = jb + i;
    int c0 = off0 + (j >> 5), c1 = off1 + (j >> 4), c2 = off2 + (j >> 3);
    int c3 = off3 + (j >> 2), c4 = off4 + (j >> 1), c5 = off5 + j;
    float P = 1.0f;
    P *= 1.0f / (1.0f + __expf(gr[c0 ^ 1] - gr[c0]));
    P *= 1.0f / (1.0f + __expf(gr[c1 ^ 1] - gr[c1]));
    P *= 1.0f / (1.0f + __expf(gr[c2 ^ 1] - gr[c2]));
    P *= 1.0f / (1.0f + __expf(gr[c3 ^ 1] - gr[c3]));
    P *= 1.0f / (1.0f + __expf(gr[c4 ^ 1] - gr[c4]));
    P *= 1.0f / (1.0f + __expf(gr[c5 ^ 1] - gr[c5]));
    orow[j] = P;
  }
}

extern "C" void kernel_launch(void* const* d_in, const int* in_sizes, int n_in,
                              void* d_out, int out_size, void* d_ws, size_t ws_size,
                              hipStream_t stream) {
  const float* x  = (const float*)d_in[0];
  const float* t0 = (const float*)d_in[1];
  const float* t1 = (const float*)d_in[2];
  const float* t2 = (const float*)d_in[3];
  const float* t3 = (const float*)d_in[4];
  const float* t4 = (const float*)d_in[5];
  float* out = (float*)d_out;
  float2* wfrag = (float2*)d_ws;            // needs 128 KB of d_ws

  // 1) build packed weights in WMMA fragment layout (16384 float2)
  build_weights_kernel<<<64, 256, 0, stream>>>(t0, t1, t2, t3, t4, wfrag);

  // 2) fused GEMM + softmax-tree kernel: 256 rows per block
  int B = in_sizes[0] / 128;                // 131072
  int grid = B / 256;                       // 512
  beam_tree_kernel<<<grid, 512, LDS_BYTES, stream>>>(x, out, wfrag);
}